// DDHGRCNnGL_22316650070466
// MI455X (gfx1250) — compile-verified
//
#include <hip/hip_runtime.h>
#include <math.h>

#define Bb   64
#define Nn   256
#define Ee   4096
#define FIN  512
#define HIDN 1024

typedef __attribute__((ext_vector_type(2))) float v2f;
typedef __attribute__((ext_vector_type(8))) float v8f;

// ---------------------------------------------------------------------------
// Weight pre-pack: W (KxN row-major) -> Wp (K/2, N, 2) so that the two K
// values of a WMMA B-fragment are adjacent in memory (single b64 load).
// ---------------------------------------------------------------------------
__global__ void pack_w_k(const float* __restrict__ W, float* __restrict__ Wp,
                         int K, int Nc) {
  const int i = blockIdx.x * blockDim.x + threadIdx.x;   // over (K/2)*Nc pairs
  const int total = (K / 2) * Nc;
  if (i >= total) return;
  const int n = i % Nc, kp = i / Nc;
  Wp[(size_t)i * 2 + 0] = W[(size_t)(2 * kp + 0) * Nc + n];
  Wp[(size_t)i * 2 + 1] = W[(size_t)(2 * kp + 1) * Nc + n];
}

// ---------------------------------------------------------------------------
// WMMA dual GEMM: OUT[m,n] = A1[m,:]@W1[:,n] + A2[m,:]@W2[:,n] + bias[n]
// A1,A2: MxK row-major ; Wp1,Wp2: packed (K/2,N,2) ; OUT: MxN row-major.
// One wave computes a 16(M) x 64(N) tile: 4 accumulator chains share each A
// fragment; fragments are software-pipelined (double-buffered) so the 8
// v_wmma per k-step overlap with the next step's 10 b64 loads.
// fp32 WMMA 16x16x4 keeps exact fp32 semantics of the reference.
// ---------------------------------------------------------------------------
__global__ __launch_bounds__(32) void wmma_dual_gemm(
    const float* __restrict__ A1, const float* __restrict__ Wp1,
    const float* __restrict__ A2, const float* __restrict__ Wp2,
    const float* __restrict__ bias, float* __restrict__ OUT,
    int M, int K, int Nc) {
  const int tileN = blockIdx.x;          // 64-wide N tile
  const int tileM = blockIdx.y;          // 16-wide M tile
  const int lane  = threadIdx.x;
  const int l15   = lane & 15;
  const int khalf = lane >> 4;           // 0 or 1
  const int mrow  = tileM * 16 + l15;    // A row for this lane
  const int ncol0 = tileN * 64 + l15;    // first B col for this lane

  const float* pa1 = A1 + (size_t)mrow * K + 2 * khalf;
  const float* pa2 = A2 + (size_t)mrow * K + 2 * khalf;
  const float* pb1 = Wp1 + ((size_t)khalf * Nc + ncol0) * 2;
  const float* pb2 = Wp2 + ((size_t)khalf * Nc + ncol0) * 2;
  const size_t bstep = (size_t)4 * Nc;   // floats per k0 += 4 (2 K-pairs)

  v8f acc0 = {}, acc1 = {}, acc2 = {}, acc3 = {};

  // preload k0 = 0 fragments
  v2f a1c = *(const v2f*)pa1;
  v2f a2c = *(const v2f*)pa2;
  v2f b1c0 = *(const v2f*)(pb1 +  0), b1c1 = *(const v2f*)(pb1 + 32);
  v2f b1c2 = *(const v2f*)(pb1 + 64), b1c3 = *(const v2f*)(pb1 + 96);
  v2f b2c0 = *(const v2f*)(pb2 +  0), b2c1 = *(const v2f*)(pb2 + 32);
  v2f b2c2 = *(const v2f*)(pb2 + 64), b2c3 = *(const v2f*)(pb2 + 96);

  for (int k0 = 0; k0 < K - 4; k0 += 4) {
    // per-lane prefetch of the A streams ~16 k-steps ahead
    if (k0 + 68 < K) {
      __builtin_prefetch(pa1 + 64, 0, 0);
      __builtin_prefetch(pa2 + 64, 0, 0);
    }
    // issue next-step loads (double buffer)
    pa1 += 4; pa2 += 4; pb1 += bstep; pb2 += bstep;
    const v2f a1n = *(const v2f*)pa1;
    const v2f a2n = *(const v2f*)pa2;
    const v2f b1n0 = *(const v2f*)(pb1 +  0), b1n1 = *(const v2f*)(pb1 + 32);
    const v2f b1n2 = *(const v2f*)(pb1 + 64), b1n3 = *(const v2f*)(pb1 + 96);
    const v2f b2n0 = *(const v2f*)(pb2 +  0), b2n1 = *(const v2f*)(pb2 + 32);
    const v2f b2n2 = *(const v2f*)(pb2 + 64), b2n3 = *(const v2f*)(pb2 + 96);

    // 8 WMMAs against registers loaded last iteration
    acc0 = __builtin_amdgcn_wmma_f32_16x16x4_f32(false, a1c, false, b1c0, (short)0, acc0, false, false);
    acc1 = __builtin_amdgcn_wmma_f32_16x16x4_f32(false, a1c, false, b1c1, (short)0, acc1, false, false);
    acc2 = __builtin_amdgcn_wmma_f32_16x16x4_f32(false, a1c, false, b1c2, (short)0, acc2, false, false);
    acc3 = __builtin_amdgcn_wmma_f32_16x16x4_f32(false, a1c, false, b1c3, (short)0, acc3, false, false);
    acc0 = __builtin_amdgcn_wmma_f32_16x16x4_f32(false, a2c, false, b2c0, (short)0, acc0, false, false);
    acc1 = __builtin_amdgcn_wmma_f32_16x16x4_f32(false, a2c, false, b2c1, (short)0, acc1, false, false);
    acc2 = __builtin_amdgcn_wmma_f32_16x16x4_f32(false, a2c, false, b2c2, (short)0, acc2, false, false);
    acc3 = __builtin_amdgcn_wmma_f32_16x16x4_f32(false, a2c, false, b2c3, (short)0, acc3, false, false);

    a1c = a1n; a2c = a2n;
    b1c0 = b1n0; b1c1 = b1n1; b1c2 = b1n2; b1c3 = b1n3;
    b2c0 = b2n0; b2c1 = b2n1; b2c2 = b2n2; b2c3 = b2n3;
  }

  // epilogue: final k-step
  acc0 = __builtin_amdgcn_wmma_f32_16x16x4_f32(false, a1c, false, b1c0, (short)0, acc0, false, false);
  acc1 = __builtin_amdgcn_wmma_f32_16x16x4_f32(false, a1c, false, b1c1, (short)0, acc1, false, false);
  acc2 = __builtin_amdgcn_wmma_f32_16x16x4_f32(false, a1c, false, b1c2, (short)0, acc2, false, false);
  acc3 = __builtin_amdgcn_wmma_f32_16x16x4_f32(false, a1c, false, b1c3, (short)0, acc3, false, false);
  acc0 = __builtin_amdgcn_wmma_f32_16x16x4_f32(false, a2c, false, b2c0, (short)0, acc0, false, false);
  acc1 = __builtin_amdgcn_wmma_f32_16x16x4_f32(false, a2c, false, b2c1, (short)0, acc1, false, false);
  acc2 = __builtin_amdgcn_wmma_f32_16x16x4_f32(false, a2c, false, b2c2, (short)0, acc2, false, false);
  acc3 = __builtin_amdgcn_wmma_f32_16x16x4_f32(false, a2c, false, b2c3, (short)0, acc3, false, false);

  const int rbase = tileM * 16 + khalf * 8;
  v8f accs[4] = {acc0, acc1, acc2, acc3};
#pragma unroll
  for (int ti = 0; ti < 4; ti++) {
    const int ccol = ncol0 + ti * 16;
    const float bval = bias[ccol];
#pragma unroll
    for (int j = 0; j < 8; j++)
      OUT[(size_t)(rbase + j) * Nc + ccol] = accs[ti][j] + bval;
  }
}

// ---------------------------------------------------------------------------
// Edge gather + segment-sum via atomics.
// ---------------------------------------------------------------------------
__global__ void gather_scatter_k(const float* __restrict__ x,
                                 const int* __restrict__ src,
                                 const int* __restrict__ dst, int bstride,
                                 const float* __restrict__ emask,
                                 float* __restrict__ aggr, int nNodes, int C) {
  const int e = blockIdx.x;
  const int b = blockIdx.y;
  const float m = emask ? emask[(size_t)b * Ee + e] : 1.0f;
  if (m == 0.0f) return;
  const int s = src[(size_t)b * bstride + e];
  const int d = dst[(size_t)b * bstride + e];
  const float* xs = x + ((size_t)b * nNodes + s) * C;
  float* ag = aggr + ((size_t)b * nNodes + d) * C;
  for (int f = threadIdx.x; f < C; f += blockDim.x)
    atomicAdd(&ag[f], xs[f] * m);
}

// ---------------------------------------------------------------------------
// Column mean/var over R rows with stride C (population variance).
// ---------------------------------------------------------------------------
__global__ void colstats_k(const float* __restrict__ h, float* __restrict__ mean,
                           float* __restrict__ var, int R, int C) {
  const int c = blockIdx.x;
  float s = 0.f, s2 = 0.f;
  for (int r = threadIdx.x; r < R; r += blockDim.x) {
    const float v = h[(size_t)r * C + c];
    s += v; s2 += v * v;
  }
  __shared__ float sh[256], sh2[256];
  sh[threadIdx.x] = s; sh2[threadIdx.x] = s2;
  __syncthreads();
  for (int off = 128; off > 0; off >>= 1) {
    if ((int)threadIdx.x < off) {
      sh[threadIdx.x]  += sh[threadIdx.x + off];
      sh2[threadIdx.x] += sh2[threadIdx.x + off];
    }
    __syncthreads();
  }
  if (threadIdx.x == 0) {
    const float m = sh[0] / (float)R;
    mean[c] = m;
    var[c]  = sh2[0] / (float)R - m * m;
  }
}

__global__ void bn_relu_k(float* __restrict__ h, const float* __restrict__ mean,
                          const float* __restrict__ var, const float* __restrict__ g,
                          const float* __restrict__ bt, int total, int C, int do_relu) {
  const int i = blockIdx.x * blockDim.x + threadIdx.x;
  if (i >= total) return;
  const int c = i % C;
  float v = g[c] * (h[i] - mean[c]) * rsqrtf(var[c] + 1e-5f) + bt[c];
  h[i] = do_relu ? fmaxf(v, 0.f) : v;
}

__global__ void vec_norm_k(const float* __restrict__ w, float* __restrict__ out, int n) {
  __shared__ float sh[256];
  float s = 0.f;
  for (int i = threadIdx.x; i < n; i += 256) s += w[i] * w[i];
  sh[threadIdx.x] = s;
  __syncthreads();
  for (int off = 128; off > 0; off >>= 1) {
    if ((int)threadIdx.x < off) sh[threadIdx.x] += sh[threadIdx.x + off];
    __syncthreads();
  }
  if (threadIdx.x == 0) out[0] = sqrtf(sh[0]);
}

// ---------------------------------------------------------------------------
// top-k pool: one block per batch, blockDim == n (128 or 256), k = n/2.
// Stable descending bitonic sort (ties -> lower index), matches lax.top_k.
// ---------------------------------------------------------------------------
__global__ void topk_pool_k(const float* __restrict__ h, const float* __restrict__ w,
                            const float* __restrict__ wnorm, float* __restrict__ newx,
                            int* __restrict__ inv, int n, int C) {
  const int b = blockIdx.x;
  const int t = threadIdx.x;
  __shared__ float sc[256];
  __shared__ int   si[256];

  const float* hr = h + ((size_t)b * n + t) * C;
  float dot = 0.f;
  for (int c = 0; c < C; c++) dot += hr[c] * w[c];
  sc[t] = tanhf(dot / wnorm[0]);
  si[t] = t;
  __syncthreads();

  for (int ksz = 2; ksz <= n; ksz <<= 1) {
    for (int j = ksz >> 1; j > 0; j >>= 1) {
      const int ixj = t ^ j;
      if (ixj > t) {
        const bool desc = ((t & ksz) == 0);
        const float a = sc[t], bb = sc[ixj];
        const int ia = si[t], ib = si[ixj];
        const bool tLess = (a < bb) || (a == bb && ia > ib);
        const bool doswap = desc ? tLess : !tLess;
        if (doswap) { sc[t] = bb; sc[ixj] = a; si[t] = ib; si[ixj] = ia; }
      }
      __syncthreads();
    }
  }

  inv[(size_t)b * n + t] = -1;
  __syncthreads();
  const int k = n >> 1;
  if (t < k) {
    inv[(size_t)b * n + si[t]] = t;
    const float v = sc[t];
    const float* hs = h + ((size_t)b * n + si[t]) * C;
    float* o = newx + ((size_t)b * k + t) * C;
    for (int c = 0; c < C; c++) o[c] = hs[c] * v;
  }
}

__global__ void remap_edges_k(const int* __restrict__ src, const int* __restrict__ dst,
                              int bstride, const float* __restrict__ emask_in,
                              const int* __restrict__ inv, int n,
                              int* __restrict__ nsrc, int* __restrict__ ndst,
                              float* __restrict__ emask_out) {
  const int i = blockIdx.x * blockDim.x + threadIdx.x;
  if (i >= Bb * Ee) return;
  const int b = i / Ee, e = i % Ee;
  const int s = src[(size_t)b * bstride + e];
  const int d = dst[(size_t)b * bstride + e];
  const int ns = inv[(size_t)b * n + s];
  const int nd = inv[(size_t)b * n + d];
  const float em = emask_in ? emask_in[i] : 1.0f;
  const bool valid = (ns >= 0) && (nd >= 0) && (em > 0.5f);
  nsrc[i] = valid ? ns : 0;
  ndst[i] = valid ? nd : 0;
  emask_out[i] = valid ? 1.0f : 0.0f;
}

__global__ void mean_nodes_k(const float* __restrict__ h, float* __restrict__ out,
                             int n, int C) {
  const int i = blockIdx.x * blockDim.x + threadIdx.x;
  if (i >= Bb * C) return;
  const int b = i / C, c = i % C;
  const float* p = h + (size_t)b * n * C + c;
  float s = 0.f;
  for (int r = 0; r < n; r++) s += p[(size_t)r * C];
  out[i] = s / (float)n;
}

__global__ void add_vec_k(const float* a, const float* b, float* o, int total) {
  const int i = blockIdx.x * blockDim.x + threadIdx.x;
  if (i < total) o[i] = a[i] + b[i];
}

// out[b,c] = max(in[b,2c], in[b,2c+1]) ; in is (B, 2C)
__global__ void pairmax_k(const float* __restrict__ in, float* __restrict__ out, int C) {
  const int i = blockIdx.x * blockDim.x + threadIdx.x;
  if (i >= Bb * C) return;
  const int b = i / C, c = i % C;
  const float* r = in + (size_t)b * 2 * C;
  out[i] = fmaxf(r[2 * c], r[2 * c + 1]);
}

// Direct conv (VALID), weight OIHW, + bias, optional ReLU.
__global__ void conv2d_k(const float* __restrict__ in, const float* __restrict__ w,
                         const float* __restrict__ bias, float* __restrict__ out,
                         int Cin, int Cout, int Hin, int Win, int Hout, int Wout,
                         int ksz, int stride, int do_relu) {
  const int i = blockIdx.x * blockDim.x + threadIdx.x;
  const int total = Bb * Cout * Hout * Wout;
  if (i >= total) return;
  int wo = i % Wout; int t = i / Wout;
  int ho = t % Hout; t /= Hout;
  int co = t % Cout; int b = t / Cout;
  float acc = bias[co];
  for (int ci = 0; ci < Cin; ci++)
    for (int kh = 0; kh < ksz; kh++)
      for (int kw = 0; kw < ksz; kw++)
        acc += in[(((size_t)b * Cin + ci) * Hin + ho * stride + kh) * Win + wo * stride + kw]
             * w[(((size_t)co * Cin + ci) * ksz + kh) * ksz + kw];
  out[i] = do_relu ? fmaxf(acc, 0.f) : acc;
}

__global__ void bn2d_stats_k(const float* __restrict__ x, float* __restrict__ mean,
                             float* __restrict__ var, int C, int HW) {
  const int c = blockIdx.x;
  float s = 0.f, s2 = 0.f;
  for (int i = threadIdx.x; i < Bb * HW; i += blockDim.x) {
    const int b = i / HW, r = i % HW;
    const float v = x[((size_t)b * C + c) * HW + r];
    s += v; s2 += v * v;
  }
  __shared__ float sh[256], sh2[256];
  sh[threadIdx.x] = s; sh2[threadIdx.x] = s2;
  __syncthreads();
  for (int off = 128; off > 0; off >>= 1) {
    if ((int)threadIdx.x < off) {
      sh[threadIdx.x]  += sh[threadIdx.x + off];
      sh2[threadIdx.x] += sh2[threadIdx.x + off];
    }
    __syncthreads();
  }
  if (threadIdx.x == 0) {
    const float m = sh[0] / (float)(Bb * HW);
    mean[c] = m;
    var[c]  = sh2[0] / (float)(Bb * HW) - m * m;
  }
}

__global__ void bn2d_apply_k(float* __restrict__ x, const float* __restrict__ mean,
                             const float* __restrict__ var, const float* __restrict__ g,
                             const float* __restrict__ bt, int total, int C, int HW) {
  const int i = blockIdx.x * blockDim.x + threadIdx.x;
  if (i >= total) return;
  const int c = (i / HW) % C;
  x[i] = g[c] * (x[i] - mean[c]) * rsqrtf(var[c] + 1e-5f) + bt[c];
}

// z (B,1,Hout,Wout) += maxpool3x3s2(cut (B,1,Hin,Win))
__global__ void maxpool_add_k(const float* __restrict__ cut, float* __restrict__ z,
                              int Hin, int Win, int Hout, int Wout) {
  const int i = blockIdx.x * blockDim.x + threadIdx.x;
  if (i >= Bb * Hout * Wout) return;
  int wo = i % Wout; int t = i / Wout;
  int ho = t % Hout; int b = t / Hout;
  float m = -3.402823466e38f;
  for (int kh = 0; kh < 3; kh++)
    for (int kw = 0; kw < 3; kw++)
      m = fmaxf(m, cut[((size_t)b * Hin + ho * 2 + kh) * Win + wo * 2 + kw]);
  z[i] += m;
}

__global__ void fc_relu_k(const float* __restrict__ z, const float* __restrict__ W,
                          const float* __restrict__ bias, float* __restrict__ out) {
  const int i = blockIdx.x * blockDim.x + threadIdx.x;
  if (i >= Bb * 1024) return;
  const int b = i / 1024, j = i % 1024;
  float s = bias[j];
  for (int kk = 0; kk < 9; kk++) s += z[b * 9 + kk] * W[kk * 1024 + j];
  out[i] = fmaxf(s, 0.f);
}

__global__ void fc_out_k(const float* __restrict__ h, const float* __restrict__ W,
                         const float* __restrict__ bias, float* __restrict__ out) {
  const int i = blockIdx.x * blockDim.x + threadIdx.x;
  if (i >= Bb * 10) return;
  const int b = i / 10, o = i % 10;
  float s = bias[o];
  for (int j = 0; j < 1024; j++) s += h[(size_t)b * 1024 + j] * W[j * 10 + o];
  out[i] = s;
}

// ---------------------------------------------------------------------------
extern "C" void kernel_launch(void* const* d_in, const int* in_sizes, int n_in,
                              void* d_out, int out_size, void* d_ws, size_t ws_size,
                              hipStream_t stream) {
  const float* x       = (const float*)d_in[0];
  const int*   edge    = (const int*)d_in[1];
  const float* Wrel1   = (const float*)d_in[2];
  const float* Wroot1  = (const float*)d_in[3];
  const float* b1      = (const float*)d_in[4];
  const float* Wrel2   = (const float*)d_in[5];
  const float* Wroot2  = (const float*)d_in[6];
  const float* b2      = (const float*)d_in[7];
  const float* bn1_g   = (const float*)d_in[8];
  const float* bn1_b   = (const float*)d_in[9];
  const float* bn2_g   = (const float*)d_in[10];
  const float* bn2_b   = (const float*)d_in[11];
  const float* pool_w1 = (const float*)d_in[12];
  const float* pool_w2 = (const float*)d_in[13];
  const float* BN1_g   = (const float*)d_in[14];
  const float* BN1_b   = (const float*)d_in[15];
  const float* BN2_g   = (const float*)d_in[16];
  const float* BN2_b   = (const float*)d_in[17];
  const float* c1_w    = (const float*)d_in[18];
  const float* c1_b    = (const float*)d_in[19];
  const float* bn2d1_g = (const float*)d_in[20];
  const float* bn2d1_b = (const float*)d_in[21];
  const float* c2_w    = (const float*)d_in[22];
  const float* c2_b    = (const float*)d_in[23];
  const float* bn2d2_g = (const float*)d_in[24];
  const float* bn2d2_b = (const float*)d_in[25];
  const float* c3_w    = (const float*)d_in[26];
  const float* c3_b    = (const float*)d_in[27];
  const float* c4_w    = (const float*)d_in[28];
  const float* c4_b    = (const float*)d_in[29];
  const float* bn2d3_g = (const float*)d_in[30];
  const float* bn2d3_b = (const float*)d_in[31];
  const float* c5_w    = (const float*)d_in[32];
  const float* c5_b    = (const float*)d_in[33];
  const float* bn2d4_g = (const float*)d_in[34];
  const float* bn2d4_b = (const float*)d_in[35];
  const float* c6_w    = (const float*)d_in[36];
  const float* c6_b    = (const float*)d_in[37];
  const float* fc_w    = (const float*)d_in[38];
  const float* fc_b    = (const float*)d_in[39];
  const float* fc1_w   = (const float*)d_in[40];
  const float* fc1_b   = (const float*)d_in[41];

  // ---- workspace bump allocator with region reuse ----
  char* base = (char*)d_ws;
  size_t off = 0;
  auto alloc = [&](size_t bytes) -> void* {
    void* r = base + off;
    off += (bytes + 255) & ~(size_t)255;
    return r;
  };
  float* region0 = (float*)alloc((size_t)Bb * Nn * HIDN * 4);      // h1 -> h2 -> conv scratch
  float* region1 = (float*)alloc((size_t)Bb * Nn * FIN * 4);       // aggr1 -> newx1
  float* region2 = (float*)alloc((size_t)Bb * (Nn/2) * HIDN * 4);  // aggr2 -> newx2
  float* wp_rel1  = (float*)alloc((size_t)FIN * HIDN * 4);         // packed weights
  float* wp_root1 = (float*)alloc((size_t)FIN * HIDN * 4);
  float* wp_rel2  = (float*)alloc((size_t)HIDN * HIDN * 4);
  float* wp_root2 = (float*)alloc((size_t)HIDN * HIDN * 4);
  float* mean  = (float*)alloc(HIDN * 4);
  float* var   = (float*)alloc(HIDN * 4);
  float* wnorm = (float*)alloc(256);
  int*   inv   = (int*)alloc((size_t)Bb * Nn * 4);
  int*   nsrc  = (int*)alloc((size_t)Bb * Ee * 4);
  int*   ndst  = (int*)alloc((size_t)Bb * Ee * 4);
  float* emaskb = (float*)alloc((size_t)Bb * Ee * 4);
  float* x1    = (float*)alloc((size_t)Bb * HIDN * 4);
  float* x2    = (float*)alloc((size_t)Bb * HIDN * 4);
  float* zsum  = (float*)alloc((size_t)Bb * HIDN * 4);
  float* z512  = (float*)alloc((size_t)Bb * 512 * 4);
  float* z256  = (float*)alloc((size_t)Bb * 256 * 4);   // conv input + cut1
  float* z7    = (float*)alloc((size_t)Bb * 49 * 4);    // becomes cut2
  float* z3    = (float*)alloc((size_t)Bb * 9 * 4);
  float* fcbuf = (float*)alloc((size_t)Bb * 1024 * 4);

  float* h1    = region0;
  float* aggr1 = region1;
  float* newx1 = region1;   // reused after GEMM1 consumed aggr1
  float* aggr2 = region2;
  float* h2    = region0;   // reused after pool1 consumed h1
  float* newx2 = region2;   // reused after GEMM2 consumed aggr2
  // conv scratch carved from region0 (free once x1/x2 are computed)
  float* cbuf1 = region0;                                 // 64*128*256
  float* cbuf2 = cbuf1 + (size_t)Bb * 128 * 256;          // 64*64*49
  float* cbuf4 = cbuf2 + (size_t)Bb * 64 * 49 + 64;       // 64*128*49
  float* cbuf5 = cbuf4 + (size_t)Bb * 128 * 49 + 64;      // 64*64*9

  const int T = 256;
  auto nb = [](int total, int thr) { return (total + thr - 1) / thr; };

  // ===== weight packing (once per launch, deterministic) =====
  pack_w_k<<<nb((FIN / 2) * HIDN, T), T, 0, stream>>>(Wrel1, wp_rel1, FIN, HIDN);
  pack_w_k<<<nb((FIN / 2) * HIDN, T), T, 0, stream>>>(Wroot1, wp_root1, FIN, HIDN);
  pack_w_k<<<nb((HIDN / 2) * HIDN, T), T, 0, stream>>>(Wrel2, wp_rel2, HIDN, HIDN);
  pack_w_k<<<nb((HIDN / 2) * HIDN, T), T, 0, stream>>>(Wroot2, wp_root2, HIDN, HIDN);

  // ===== Layer 1 graph conv =====
  hipMemsetAsync(aggr1, 0, (size_t)Bb * Nn * FIN * 4, stream);
  gather_scatter_k<<<dim3(Ee, Bb), 128, 0, stream>>>(x, edge, edge + Ee, 2 * Ee,
                                                     nullptr, aggr1, Nn, FIN);
  wmma_dual_gemm<<<dim3(HIDN / 64, (Bb * Nn) / 16), 32, 0, stream>>>(
      aggr1, wp_rel1, x, wp_root1, b1, h1, Bb * Nn, FIN, HIDN);
  colstats_k<<<HIDN, T, 0, stream>>>(h1, mean, var, Bb * Nn, HIDN);
  bn_relu_k<<<nb(Bb * Nn * HIDN, T), T, 0, stream>>>(h1, mean, var, bn1_g, bn1_b,
                                                     Bb * Nn * HIDN, HIDN, 1);

  // ===== top-k pool 1 (n=256 -> k=128) =====
  vec_norm_k<<<1, T, 0, stream>>>(pool_w1, wnorm, HIDN);
  topk_pool_k<<<Bb, Nn, 0, stream>>>(h1, pool_w1, wnorm, newx1, inv, Nn, HIDN);
  remap_edges_k<<<nb(Bb * Ee, T), T, 0, stream>>>(edge, edge + Ee, 2 * Ee, nullptr,
                                                  inv, Nn, nsrc, ndst, emaskb);
  mean_nodes_k<<<nb(Bb * HIDN, T), T, 0, stream>>>(newx1, x1, Nn / 2, HIDN);

  // ===== Layer 2 graph conv =====
  hipMemsetAsync(aggr2, 0, (size_t)Bb * (Nn / 2) * HIDN * 4, stream);
  gather_scatter_k<<<dim3(Ee, Bb), 128, 0, stream>>>(newx1, nsrc, ndst, Ee,
                                                     emaskb, aggr2, Nn / 2, HIDN);
  wmma_dual_gemm<<<dim3(HIDN / 64, (Bb * (Nn / 2)) / 16), 32, 0, stream>>>(
      aggr2, wp_rel2, newx1, wp_root2, b2, h2, Bb * (Nn / 2), HIDN, HIDN);
  colstats_k<<<HIDN, T, 0, stream>>>(h2, mean, var, Bb * (Nn / 2), HIDN);
  bn_relu_k<<<nb(Bb * (Nn / 2) * HIDN, T), T, 0, stream>>>(
      h2, mean, var, bn2_g, bn2_b, Bb * (Nn / 2) * HIDN, HIDN, 1);

  // ===== top-k pool 2 (n=128 -> k=64); remapped edges are unused afterwards =====
  vec_norm_k<<<1, T, 0, stream>>>(pool_w2, wnorm, HIDN);
  topk_pool_k<<<Bb, Nn / 2, 0, stream>>>(h2, pool_w2, wnorm, newx2, inv, Nn / 2, HIDN);
  mean_nodes_k<<<nb(Bb * HIDN, T), T, 0, stream>>>(newx2, x2, Nn / 4, HIDN);

  // ===== readout MLP head =====
  add_vec_k<<<nb(Bb * HIDN, T), T, 0, stream>>>(x1, x2, zsum, Bb * HIDN);
  pairmax_k<<<nb(Bb * 512, T), T, 0, stream>>>(zsum, z512, 512);
  colstats_k<<<512, T, 0, stream>>>(z512, mean, var, Bb, 512);
  bn_relu_k<<<nb(Bb * 512, T), T, 0, stream>>>(z512, mean, var, BN1_g, BN1_b,
                                               Bb * 512, 512, 1);
  pairmax_k<<<nb(Bb * 256, T), T, 0, stream>>>(z512, z256, 256);
  colstats_k<<<256, T, 0, stream>>>(z256, mean, var, Bb, 256);
  bn_relu_k<<<nb(Bb * 256, T), T, 0, stream>>>(z256, mean, var, BN2_g, BN2_b,
                                               Bb * 256, 256, 1);

  // ===== conv block 1 (input z256 = (B,1,16,16), also cut1) =====
  conv2d_k<<<nb(Bb * 128 * 256, T), T, 0, stream>>>(z256, c1_w, c1_b, cbuf1,
                                                    1, 128, 16, 16, 16, 16, 1, 1, 1);
  bn2d_stats_k<<<128, T, 0, stream>>>(cbuf1, mean, var, 128, 256);
  bn2d_apply_k<<<nb(Bb * 128 * 256, T), T, 0, stream>>>(cbuf1, mean, var, bn2d1_g,
                                                        bn2d1_b, Bb * 128 * 256, 128, 256);
  conv2d_k<<<nb(Bb * 64 * 49, T), T, 0, stream>>>(cbuf1, c2_w, c2_b, cbuf2,
                                                  128, 64, 16, 16, 7, 7, 3, 2, 1);
  bn2d_stats_k<<<64, T, 0, stream>>>(cbuf2, mean, var, 64, 49);
  bn2d_apply_k<<<nb(Bb * 64 * 49, T), T, 0, stream>>>(cbuf2, mean, var, bn2d2_g,
                                                      bn2d2_b, Bb * 64 * 49, 64, 49);
  conv2d_k<<<nb(Bb * 49, T), T, 0, stream>>>(cbuf2, c3_w, c3_b, z7,
                                             64, 1, 7, 7, 7, 7, 1, 1, 1);
  maxpool_add_k<<<nb(Bb * 49, T), T, 0, stream>>>(z256, z7, 16, 16, 7, 7);

  // ===== conv block 2 (input z7 = (B,1,7,7), also cut2) =====
  conv2d_k<<<nb(Bb * 128 * 49, T), T, 0, stream>>>(z7, c4_w, c4_b, cbuf4,
                                                   1, 128, 7, 7, 7, 7, 1, 1, 1);
  bn2d_stats_k<<<128, T, 0, stream>>>(cbuf4, mean, var, 128, 49);
  bn2d_apply_k<<<nb(Bb * 128 * 49, T), T, 0, stream>>>(cbuf4, mean, var, bn2d3_g,
                                                       bn2d3_b, Bb * 128 * 49, 128, 49);
  conv2d_k<<<nb(Bb * 64 * 9, T), T, 0, stream>>>(cbuf4, c5_w, c5_b, cbuf5,
                                                 128, 64, 7, 7, 3, 3, 3, 2, 1);
  bn2d_stats_k<<<64, T, 0, stream>>>(cbuf5, mean, var, 64, 9);
  bn2d_apply_k<<<nb(Bb * 64 * 9, T), T, 0, stream>>>(cbuf5, mean, var, bn2d4_g,
                                                     bn2d4_b, Bb * 64 * 9, 64, 9);
  conv2d_k<<<nb(Bb * 9, T), T, 0, stream>>>(cbuf5, c6_w, c6_b, z3,
                                            64, 1, 3, 3, 3, 3, 1, 1, 1);
  maxpool_add_k<<<nb(Bb * 9, T), T, 0, stream>>>(z7, z3, 7, 7, 3, 3);

  // ===== FC head =====
  fc_relu_k<<<nb(Bb * 1024, T), T, 0, stream>>>(z3, fc_w, fc_b, fcbuf);
  fc_out_k<<<nb(Bb * 10, T), T, 0, stream>>>(fcbuf, fc1_w, fc1_b, (float*)d_out);
}